// GNN_90134183674479
// MI455X (gfx1250) — compile-verified
//
#include <hip/hip_runtime.h>

typedef __attribute__((ext_vector_type(2))) float v2f;
typedef __attribute__((ext_vector_type(8))) float v8f;

#define N_NODES   100000
#define N_EDGES   1600000
#define F         64      // IN_FEATS == NUM_CLASSES == 64

// ---------------------------------------------------------------------------
// Kernel 0: zero the workspace (agg [N,64] + deg [N])
// ---------------------------------------------------------------------------
__global__ void GNN_zero_kernel(float* __restrict__ ws, long n) {
  long i = (long)blockIdx.x * blockDim.x + threadIdx.x;
  long stride = (long)gridDim.x * blockDim.x;
  for (; i < n; i += stride) ws[i] = 0.0f;
}

// ---------------------------------------------------------------------------
// Kernel 1: edge scatter.  One wave32 per edge; lane l handles feats
// [2l, 2l+1] as a float2 gather + two global_atomic_add_f32.  Lane 0 bumps
// the in-degree.  src/dst loads are wave-uniform -> coalesce to one request.
// ---------------------------------------------------------------------------
__global__ __launch_bounds__(256)
void GNN_scatter_kernel(const float* __restrict__ in_feat,
                        const int*   __restrict__ src,
                        const int*   __restrict__ dst,
                        float* __restrict__ agg,
                        float* __restrict__ deg) {
  int lane = threadIdx.x & 31;
  int edge = (int)((blockIdx.x * (unsigned)blockDim.x + threadIdx.x) >> 5);
  if (edge >= N_EDGES) return;
  int s = src[edge];
  int d = dst[edge];
  v2f x = *(const v2f*)(in_feat + (long)s * F + 2 * lane);
  float* ap = agg + (long)d * F + 2 * lane;
  atomicAdd(ap,     x.x);
  atomicAdd(ap + 1, x.y);
  if (lane == 0) atomicAdd(deg + d, 1.0f);
}

// ---------------------------------------------------------------------------
// Kernel 2: fused  out = sigmoid(X@Wself + (agg/max(deg,1))@Wneigh + bias)
// One wave per 16x16 output tile, v_wmma_f32_16x16x4_f32, K=64 -> 16 WMMAs
// per operand matrix (32 per tile).  256 threads = 8 waves = 2 M-tiles x
// 4 N-tiles = 32 node rows per block.  100000 % 32 == 0 -> exact tiling,
// EXEC all-ones around WMMA as the ISA requires.
// ---------------------------------------------------------------------------
__global__ __launch_bounds__(256)
void GNN_wmma_kernel(const float* __restrict__ x,       // [N,64]
                     const float* __restrict__ agg,     // [N,64]
                     const float* __restrict__ deg,     // [N]
                     const float* __restrict__ Wself,   // [64,64]
                     const float* __restrict__ Wneigh,  // [64,64]
                     const float* __restrict__ bias,    // [64]
                     float* __restrict__ out) {         // [N,64]
  const int lane  = threadIdx.x & 31;
  const int wave  = threadIdx.x >> 5;        // 0..7
  const int mtile = blockIdx.x * 2 + (wave >> 2);
  const int n0    = (wave & 3) * 16;         // N-tile origin
  const int row0  = mtile * 16;              // M-tile origin
  const int lm    = lane & 15;               // M (A/C) or N (B/C/D) index
  const int hi    = lane >> 4;               // 0: K+0/K+1 half, 1: K+2/K+3 half

  int row = row0 + lm;                       // A-fragment source row
  if (row >= N_NODES) row = N_NODES - 1;     // defensive clamp (exact here)

  const float dv  = deg[row];
  const float inv = 1.0f / fmaxf(dv, 1.0f);

  const float* __restrict__ xrow = x   + (long)row * F;
  const float* __restrict__ arow = agg + (long)row * F;

  v8f acc = {};
  #pragma unroll
  for (int kk = 0; kk < F; kk += 4) {
    // ---- self term: A = X tile, B = Wself[kk..kk+3][n0..n0+15] ----
    v2f aX = *(const v2f*)(xrow + kk + 2 * hi);      // (K+2hi, K+2hi+1)
    v2f bS;
    bS.x = Wself[(kk + 2 * hi    ) * F + n0 + lm];   // VGPR0: rows K, K+2
    bS.y = Wself[(kk + 2 * hi + 1) * F + n0 + lm];   // VGPR1: rows K+1, K+3
    acc = __builtin_amdgcn_wmma_f32_16x16x4_f32(
        false, aX, false, bS, (short)0, acc, false, false);

    // ---- neighbor term: A = mean-aggregated tile, B = Wneigh ----
    v2f aH = *(const v2f*)(arow + kk + 2 * hi);
    aH.x *= inv; aH.y *= inv;
    v2f bN;
    bN.x = Wneigh[(kk + 2 * hi    ) * F + n0 + lm];
    bN.y = Wneigh[(kk + 2 * hi + 1) * F + n0 + lm];
    acc = __builtin_amdgcn_wmma_f32_16x16x4_f32(
        false, aH, false, bN, (short)0, acc, false, false);
  }

  // C/D layout: VGPR j -> M = j + 8*hi, N = lm  (ISA 7.12.2)
  const float b = bias[n0 + lm];
  #pragma unroll
  for (int j = 0; j < 8; ++j) {
    int r = row0 + j + 8 * hi;
    if (r < N_NODES) {
      float v = acc[j] + b;
      out[(long)r * F + n0 + lm] = 1.0f / (1.0f + __expf(-v));
    }
  }
}

// ---------------------------------------------------------------------------
extern "C" void kernel_launch(void* const* d_in, const int* in_sizes, int n_in,
                              void* d_out, int out_size, void* d_ws, size_t ws_size,
                              hipStream_t stream) {
  const float* in_feat = (const float*)d_in[0];
  const int*   src     = (const int*)  d_in[1];
  const int*   dst     = (const int*)  d_in[2];
  const float* Wself   = (const float*)d_in[3];
  const float* Wneigh  = (const float*)d_in[4];
  const float* bias    = (const float*)d_in[5];
  float*       out     = (float*)d_out;

  float* agg = (float*)d_ws;                       // N*64 floats (25.6 MB)
  float* deg = agg + (size_t)N_NODES * F;          // N floats

  const long zn = (long)N_NODES * F + N_NODES;
  GNN_zero_kernel<<<2048, 256, 0, stream>>>((float*)d_ws, zn);

  // one wave per edge: E*32 threads / 256 threads = E/8 blocks
  const int sblocks = (N_EDGES + 7) / 8;
  GNN_scatter_kernel<<<sblocks, 256, 0, stream>>>(in_feat, src, dst, agg, deg);

  // 32 node rows per block
  const int gblocks = (N_NODES + 31) / 32;
  GNN_wmma_kernel<<<gblocks, 256, 0, stream>>>(in_feat, agg, deg,
                                               Wself, Wneigh, bias, out);
}